// GGNN_38878043963478
// MI455X (gfx1250) — compile-verified
//
#include <hip/hip_runtime.h>

// GGNN step for MI455X (gfx1250, wave32).
//   Stage 0: zero message accumulator m[N,16] in d_ws.
//   Stage 1: per-edge msg = A_type @ h_src, scatter via global_atomic_add_f32.
//            (All 8 edge matrices = 8KB, staged in LDS per block; edge index
//             streams loaded nontemporally to protect L2-resident features.)
//   Stage 2: fused GRU via V_WMMA_F32_16X16X4_F32 (full fp32 matrix path):
//            per 16-node tile, one wave preloads all operand fragments, then
//            issues 24 back-to-back WMMAs, then gates + nontemporal stores.

#define HID 16

typedef __attribute__((ext_vector_type(2))) float v2f;
typedef __attribute__((ext_vector_type(8))) float v8f;

__device__ __forceinline__ float sigmoidf_(float x) {
    return 1.0f / (1.0f + expf(-x));
}

// ---------------- Stage 0: zero workspace accumulator ----------------
__global__ __launch_bounds__(256) void ggnn_zero_kernel(float4* __restrict__ p, int n4) {
    int i = blockIdx.x * blockDim.x + threadIdx.x;
    if (i < n4) p[i] = make_float4(0.f, 0.f, 0.f, 0.f);
}

// ---------------- Stage 1: edge message + scatter-sum ----------------
__global__ __launch_bounds__(256) void ggnn_edge_kernel(
    const float* __restrict__ features,
    const int* __restrict__ esrc,
    const int* __restrict__ edst,
    const int* __restrict__ etyp,
    const float* __restrict__ edge_emb,
    float* __restrict__ m,
    int nEdges)
{
    __shared__ float Alds[8 * 256];  // 8 edge-type matrices, 8 KB
    for (int i = threadIdx.x; i < 8 * 256; i += 256) Alds[i] = edge_emb[i];
    __syncthreads();

    int e = blockIdx.x * blockDim.x + threadIdx.x;
    if (e >= nEdges) return;

    // Single-use streams: nontemporal so they don't evict features from L2.
    int s = __builtin_nontemporal_load(esrc + e);
    int d = __builtin_nontemporal_load(edst + e);
    int t = __builtin_nontemporal_load(etyp + e);

    const float4* hp = (const float4*)(features + (size_t)s * HID);
    float4 h0 = hp[0], h1 = hp[1], h2 = hp[2], h3 = hp[3];

    const float* A = Alds + t * 256;
    float* mrow = m + (size_t)d * HID;

#pragma unroll
    for (int r = 0; r < 16; ++r) {
        const float4* ar = (const float4*)(A + r * 16);
        float4 a0 = ar[0], a1 = ar[1], a2 = ar[2], a3 = ar[3];
        float acc = a0.x * h0.x + a0.y * h0.y + a0.z * h0.z + a0.w * h0.w;
        acc += a1.x * h1.x + a1.y * h1.y + a1.z * h1.z + a1.w * h1.w;
        acc += a2.x * h2.x + a2.y * h2.y + a2.z * h2.z + a2.w * h2.w;
        acc += a3.x * h3.x + a3.y * h3.y + a3.z * h3.z + a3.w * h3.w;
        atomicAdd(mrow + r, acc);
    }
}

// ---------------- Stage 2: fused GRU via WMMA f32 ----------------
// One wave (32 lanes) per tile of 16 nodes.
// A layout (16x4 f32, 2 VGPRs): lane L -> row m = L&15, k = 2*(L>>4) + elem.
// B layout (4x16 f32, 2 VGPRs): lane L -> col n = L&15, k = 2*(L>>4) + elem.
// C/D layout (16x16 f32, 8 VGPRs): lane L, elem v -> row v + 8*(L>>4), col L&15.
__global__ __launch_bounds__(256) void ggnn_gru_wmma_kernel(
    const float* __restrict__ m,
    const float* __restrict__ h,
    const float* __restrict__ w_ih,   // [48,16] row-major
    const float* __restrict__ w_hh,   // [48,16] row-major
    const float* __restrict__ b_ih,   // [48]
    const float* __restrict__ b_hh,   // [48]
    float* __restrict__ out,
    int nTiles)
{
    const int wave = threadIdx.x >> 5;
    const int lane = threadIdx.x & 31;
    const int tile = blockIdx.x * (blockDim.x >> 5) + wave;
    if (tile >= nTiles) return;          // whole-wave uniform exit: EXEC all-1s below

    const int half = lane >> 4;          // 0 or 1
    const int l16  = lane & 15;
    const int nodeBase = tile * 16;

    // ---- Preload ALL operand fragments up front (clustered loads, one wait) ----
    v2f a_m[4], a_h[4];          // A fragments for m-tile and h-tile
    v2f b_i[3][4], b_h[3][4];    // B fragments for w_ih^T / w_hh^T column blocks
#pragma unroll
    for (int kc = 0; kc < 4; ++kc) {
        const int kcol = kc * 4 + half * 2;  // even -> 8B-aligned v2f loads
        a_m[kc] = *(const v2f*)(m + (size_t)(nodeBase + l16) * HID + kcol);
        a_h[kc] = *(const v2f*)(h + (size_t)(nodeBase + l16) * HID + kcol);
#pragma unroll
        for (int c = 0; c < 3; ++c) {
            // B[k][n] = w[16c + n][k]; contiguous in k within a row of w.
            b_i[c][kc] = *(const v2f*)(w_ih + (size_t)(c * 16 + l16) * HID + kcol);
            b_h[c][kc] = *(const v2f*)(w_hh + (size_t)(c * 16 + l16) * HID + kcol);
        }
    }
    // Prefetch h values needed by the gate combine (C/D layout positions).
    float hv[8];
#pragma unroll
    for (int v = 0; v < 8; ++v)
        hv[v] = h[(size_t)(nodeBase + v + 8 * half) * HID + l16];

    // Accumulators initialized with broadcast biases: C[node][dim] = b[16c + dim]
    v8f gi[3], gh[3];
#pragma unroll
    for (int c = 0; c < 3; ++c) {
        float bi = b_ih[c * 16 + l16];
        float bh = b_hh[c * 16 + l16];
        v8f vi = {bi, bi, bi, bi, bi, bi, bi, bi};
        v8f vh = {bh, bh, bh, bh, bh, bh, bh, bh};
        gi[c] = vi;
        gh[c] = vh;
    }

    // ---- 24 back-to-back WMMAs; each accumulator chain has distance-6 deps ----
#pragma unroll
    for (int kc = 0; kc < 4; ++kc) {
#pragma unroll
        for (int c = 0; c < 3; ++c) {
            gi[c] = __builtin_amdgcn_wmma_f32_16x16x4_f32(
                        false, a_m[kc], false, b_i[c][kc], (short)0, gi[c], false, false);
            gh[c] = __builtin_amdgcn_wmma_f32_16x16x4_f32(
                        false, a_h[kc], false, b_h[c][kc], (short)0, gh[c], false, false);
        }
    }

    // ---- Gates + output, directly in accumulator layout ----
#pragma unroll
    for (int v = 0; v < 8; ++v) {
        const int node = nodeBase + v + 8 * half;
        const float r = sigmoidf_(gi[0][v] + gh[0][v]);
        const float z = sigmoidf_(gi[1][v] + gh[1][v]);
        const float n = tanhf(gi[2][v] + r * gh[2][v]);
        // Write-only result: nontemporal store.
        __builtin_nontemporal_store((1.0f - z) * n + z * hv[v],
                                    out + (size_t)node * HID + l16);
    }
}

extern "C" void kernel_launch(void* const* d_in, const int* in_sizes, int n_in,
                              void* d_out, int out_size, void* d_ws, size_t ws_size,
                              hipStream_t stream) {
    const float* features = (const float*)d_in[0];
    const int*   esrc     = (const int*)d_in[1];
    const int*   edst     = (const int*)d_in[2];
    const int*   etyp     = (const int*)d_in[3];
    const float* edge_emb = (const float*)d_in[4];
    const float* w_ih     = (const float*)d_in[5];
    const float* w_hh     = (const float*)d_in[6];
    const float* b_ih     = (const float*)d_in[7];
    const float* b_hh     = (const float*)d_in[8];
    float* out = (float*)d_out;
    float* m   = (float*)d_ws;   // [nNodes,16] message accumulator (6.4 MB)

    const int nNodes = in_sizes[0] / HID;
    const int nEdges = in_sizes[1];

    // Stage 0: zero accumulator (harness poisons d_ws).
    const int n4 = (nNodes * HID) / 4;
    ggnn_zero_kernel<<<(n4 + 255) / 256, 256, 0, stream>>>((float4*)m, n4);

    // Stage 1: edge messages + atomic scatter-sum.
    ggnn_edge_kernel<<<(nEdges + 255) / 256, 256, 0, stream>>>(
        features, esrc, edst, etyp, edge_emb, m, nEdges);

    // Stage 2: fused GRU (WMMA f32), one wave per 16-node tile, 8 waves/block.
    const int nTiles = (nNodes + 15) / 16;
    ggnn_gru_wmma_kernel<<<(nTiles + 7) / 8, 256, 0, stream>>>(
        m, features, w_ih, w_hh, b_ih, b_hh, out, nTiles);
}